// DistanceAwareMultiheadAttention_9990093931120
// MI455X (gfx1250) — compile-verified
//
#include <hip/hip_runtime.h>
#include <hip/hip_bf16.h>
#include <cstdint>

// ---------------------------------------------------------------------------
// DistanceAwareMultiheadAttention, MI455X (gfx1250), wave32 + WMMA fp32 path
// + async global->LDS double-buffered K/V staging (ASYNCcnt).
// B=4 S=1024 E=512 H=8 D=64
// ---------------------------------------------------------------------------

#define B_ 4
#define S_ 1024
#define E_ 512
#define H_ 8
#define D_ 64
#define MAXD_INV (1.0f / 141421.356237f)   // 1/(100000*sqrt(2))

typedef __attribute__((ext_vector_type(2))) float v2f;
typedef __attribute__((ext_vector_type(8))) float v8f;

__device__ __forceinline__ v8f wmma4(v2f a, v2f b, v8f c) {
    // V_WMMA_F32_16X16X4_F32 : D(16x16,f32) = A(16x4,f32) * B(4x16,f32) + C
    return __builtin_amdgcn_wmma_f32_16x16x4_f32(
        false, a, false, b, (short)0, c, false, false);
}

// Async global -> LDS, 16 bytes per lane, tracked by ASYNCcnt (gfx1250).
__device__ __forceinline__ void async_b128_to_lds(const float* gptr, float* lptr) {
    unsigned lofs = (unsigned)(uintptr_t)lptr;        // flat LDS addr: low 32 bits = LDS offset
    unsigned long long ga = (unsigned long long)(uintptr_t)gptr;
    asm volatile("global_load_async_to_lds_b128 %0, %1, off"
                 :: "v"(lofs), "v"(ga) : "memory");
}

// ---------------------------------------------------------------------------
// Kernel 1: fused Q/K/V projection  Out = X @ W^T + b
//   M = B*S = 4096, N = E = 512, K = E = 512, fp32 WMMA 16x16x4
//   One wave computes a 32x64 output tile (2 row-tiles x 4 col-tiles).
//   blockIdx.z in {0,1,2} selects Q / K / V.
// ---------------------------------------------------------------------------
__global__ __launch_bounds__(128) void qkv_proj_kernel(
    const float* __restrict__ q_in, const float* __restrict__ k_in,
    const float* __restrict__ v_in,
    const float* __restrict__ Wq, const float* __restrict__ Wk,
    const float* __restrict__ Wv,
    const float* __restrict__ bq, const float* __restrict__ bk,
    const float* __restrict__ bv,
    float* __restrict__ qo, float* __restrict__ ko, float* __restrict__ vo)
{
    const int z = blockIdx.z;
    const float* X  = (z == 0) ? q_in : (z == 1) ? k_in : v_in;
    const float* W  = (z == 0) ? Wq   : (z == 1) ? Wk   : Wv;
    const float* Bb = (z == 0) ? bq   : (z == 1) ? bk   : bv;
    float*       O  = (z == 0) ? qo   : (z == 1) ? ko   : vo;

    const int lane = threadIdx.x & 31;
    const int wave = threadIdx.x >> 5;
    const int tile = blockIdx.x * 4 + wave;          // 0..1023
    const int rt   = tile >> 3;                      // M/32 = 128 row tiles
    const int ct   = tile & 7;                       // N/64 = 8 col tiles
    const int rbase = rt * 32;
    const int cbase = ct * 64;
    const int hl = lane >> 4;                        // half-wave select
    const int ln = lane & 15;

    // C/D layout: VGPR r, lane L -> row = r + 8*(L>>4), col = L&15
    v8f acc[2][4];
    #pragma unroll
    for (int ch = 0; ch < 4; ++ch) {
        float bias = Bb[cbase + ch * 16 + ln];
        v8f t;
        #pragma unroll
        for (int r = 0; r < 8; ++r) t[r] = bias;
        acc[0][ch] = t; acc[1][ch] = t;
    }

    const float* a0p = X + (size_t)(rbase + ln)      * E_;
    const float* a1p = X + (size_t)(rbase + 16 + ln) * E_;
    const float* wp0 = W + (size_t)(cbase +  0 + ln) * E_;
    const float* wp1 = W + (size_t)(cbase + 16 + ln) * E_;
    const float* wp2 = W + (size_t)(cbase + 32 + ln) * E_;
    const float* wp3 = W + (size_t)(cbase + 48 + ln) * E_;

    #pragma unroll 4
    for (int k0 = 0; k0 < E_; k0 += 4) {
        const int kb = k0 + hl * 2;                  // lane<16: K0,K1 ; lane>=16: K2,K3
        v2f a0 = *(const v2f*)(a0p + kb);
        v2f a1 = *(const v2f*)(a1p + kb);
        v2f b0 = *(const v2f*)(wp0 + kb);
        v2f b1 = *(const v2f*)(wp1 + kb);
        v2f b2 = *(const v2f*)(wp2 + kb);
        v2f b3 = *(const v2f*)(wp3 + kb);
        acc[0][0] = wmma4(a0, b0, acc[0][0]);
        acc[1][0] = wmma4(a1, b0, acc[1][0]);
        acc[0][1] = wmma4(a0, b1, acc[0][1]);
        acc[1][1] = wmma4(a1, b1, acc[1][1]);
        acc[0][2] = wmma4(a0, b2, acc[0][2]);
        acc[1][2] = wmma4(a1, b2, acc[1][2]);
        acc[0][3] = wmma4(a0, b3, acc[0][3]);
        acc[1][3] = wmma4(a1, b3, acc[1][3]);
    }

    #pragma unroll
    for (int i = 0; i < 2; ++i)
        #pragma unroll
        for (int ch = 0; ch < 4; ++ch)
            #pragma unroll
            for (int r = 0; r < 8; ++r) {
                int row = rbase + i * 16 + r + 8 * hl;
                int col = cbase + ch * 16 + ln;
                O[(size_t)row * E_ + col] = acc[i][ch][r];
            }
}

// ---------------------------------------------------------------------------
// Kernel 2: per-(b,h,s) scalar edge dots
//   a0 = q . Ek[0], a1 = q . Ek[1], c0 = k . Eq[0], c1 = k . Eq[1]
// ---------------------------------------------------------------------------
__global__ __launch_bounds__(256) void edge_dots_kernel(
    const float* __restrict__ qo, const float* __restrict__ ko,
    const float* __restrict__ Ek, const float* __restrict__ Eq,
    float* __restrict__ a0, float* __restrict__ a1,
    float* __restrict__ c0, float* __restrict__ c1)
{
    int idx = blockIdx.x * 256 + threadIdx.x;        // 0 .. B*H*S-1
    if (idx >= B_ * H_ * S_) return;
    int s  = idx & (S_ - 1);
    int bh = idx >> 10;
    int h  = bh & (H_ - 1);
    int b  = bh >> 3;
    const float* qrow = qo + ((size_t)(b * S_ + s) * E_ + h * D_);
    const float* krow = ko + ((size_t)(b * S_ + s) * E_ + h * D_);
    float sa0 = 0.f, sa1 = 0.f, sc0 = 0.f, sc1 = 0.f;
    #pragma unroll 8
    for (int d = 0; d < D_; ++d) {
        float qv = qrow[d], kv = krow[d];
        sa0 += qv * Ek[d];      sa1 += qv * Ek[D_ + d];
        sc0 += kv * Eq[d];      sc1 += kv * Eq[D_ + d];
    }
    a0[idx] = sa0; a1[idx] = sa1; c0[idx] = sc0; c1[idx] = sc1;
}

// ---------------------------------------------------------------------------
// Kernel 3: flash-style distance-aware attention.
//   grid = (S/64, B*H); block = 128 (4 waves). Each wave owns a 16-row query
//   tile; block streams 16-key tiles through double-buffered LDS staged with
//   GLOBAL_LOAD_ASYNC_TO_LDS_B128 (ASYNCcnt), overlapping the next tile's
//   global traffic with the current tile's WMMA + softmax.
//   All logit scales pre-baked by 1/sqrt(D)=1/8:  q *= 0.25 (=> 2qk/8),
//   bias terms *= 0.125.
// ---------------------------------------------------------------------------
#define LK 68    // padded LDS row stride for 16x64 K/V tiles (bank-conflict free)
#define LP 17    // padded LDS row stride for 16x16 P tiles
#define NKT (S_ / 16)

__global__ __launch_bounds__(128) void attn_kernel(
    const float* __restrict__ qo, const float* __restrict__ ko,
    const float* __restrict__ vo, const float* __restrict__ pos,
    const float* __restrict__ a0g, const float* __restrict__ a1g,
    const float* __restrict__ c0g, const float* __restrict__ c1g,
    const float* __restrict__ Ev, float* __restrict__ out)
{
    __shared__ __align__(16) float kT[2][16 * LK];
    __shared__ __align__(16) float vT[2][16 * LK];
    __shared__ float pT[4][16 * LP];
    __shared__ float kx[2][16], ky[2][16], c0s[2][16], c1s[2][16];

    const int bh = blockIdx.y;
    const int h  = bh & (H_ - 1);
    const int b  = bh >> 3;
    const int lane = threadIdx.x & 31;
    const int wave = threadIdx.x >> 5;
    const int hl = lane >> 4, ln = lane & 15;
    const int qbase = blockIdx.x * 64 + wave * 16;

    // Issue async staging of key-tile `kt` into LDS buffer `buf`.
    auto issue_tile = [&](int kt, int buf) {
        const int t   = threadIdx.x;
        const int kb0 = kt * 16;
        #pragma unroll
        for (int half = 0; half < 2; ++half) {
            int le  = t * 4 + half * 512;             // 1024 floats per matrix
            int row = le >> 6, col = le & 63;
            size_t g = (size_t)(b * S_ + kb0 + row) * E_ + h * D_ + col;
            async_b128_to_lds(ko + g, &kT[buf][row * LK + col]);
            async_b128_to_lds(vo + g, &vT[buf][row * LK + col]);
        }
        if (t < 16) {
            kx[buf][t] = pos[((size_t)b * S_ + kb0 + t) * 2 + 0];
            ky[buf][t] = pos[((size_t)b * S_ + kb0 + t) * 2 + 1];
        } else if (t >= 32 && t < 48) {
            c0s[buf][t - 32] = c0g[(size_t)bh * S_ + kb0 + (t - 32)] * 0.125f;
        } else if (t >= 64 && t < 80) {
            c1s[buf][t - 64] = c1g[(size_t)bh * S_ + kb0 + (t - 64)] * 0.125f;
        }
    };

    // --- Q fragments for WMMA A-matrix (16x4 steps over K=64), prescaled ---
    v2f qf[16];
    {
        const float* qrow = qo + ((size_t)(b * S_ + qbase + ln) * E_ + h * D_);
        #pragma unroll
        for (int s = 0; s < 16; ++s) {
            int kd = 4 * s + hl * 2;
            v2f t = *(const v2f*)(qrow + kd);
            qf[s].x = t.x * 0.25f;
            qf[s].y = t.y * 0.25f;
        }
    }

    // --- per-row state (rows r and r+8 live in the two half-waves) ---
    float a0r[8], a1r[8], qxr[8], qyr[8], m[8], l[8], t0[8];
    v8f Oacc[4];
    #pragma unroll
    for (int ch = 0; ch < 4; ++ch)
        #pragma unroll
        for (int r = 0; r < 8; ++r) Oacc[ch][r] = 0.f;
    #pragma unroll
    for (int r = 0; r < 8; ++r) {
        int row = qbase + r + 8 * hl;
        a0r[r] = a0g[(size_t)bh * S_ + row] * 0.125f;
        a1r[r] = a1g[(size_t)bh * S_ + row] * 0.125f;
        qxr[r] = pos[((size_t)b * S_ + row) * 2 + 0];
        qyr[r] = pos[((size_t)b * S_ + row) * 2 + 1];
        m[r] = -1e30f; l[r] = 0.f; t0[r] = 0.f;
    }

    // ---- software pipeline: prefetch tile 0, then overlap ----
    issue_tile(0, 0);
    int cur = 0;

    for (int kt = 0; kt < NKT; ++kt) {
        // our async loads into buffer `cur` must have landed in LDS...
        asm volatile("s_wait_asynccnt 0x0" ::: "memory");
        // ...and everyone else's too; also closes WAR on buffer cur^1.
        __syncthreads();
        if (kt + 1 < NKT) issue_tile(kt + 1, cur ^ 1);

        // ---- S-tile = (2 q k^T)/8 via 16 fp32 WMMAs (K=64) ----
        v8f c;
        #pragma unroll
        for (int r = 0; r < 8; ++r) c[r] = 0.f;
        #pragma unroll
        for (int s = 0; s < 16; ++s) {
            int dd = 4 * s + hl * 2;
            v2f bfr = *(const v2f*)&kT[cur][ln * LK + dd];  // B[k=dd..dd+1][n=ln]
            c = wmma4(qf[s], bfr, c);
        }

        // ---- distance bias + online softmax (row = lanes within half-wave) ----
        const float kxj = kx[cur][ln], kyj = ky[cur][ln];
        const float c0j = c0s[cur][ln], c1j = c1s[cur][ln];
        #pragma unroll
        for (int r = 0; r < 8; ++r) {
            float dx = qxr[r] - kxj, dy = qyr[r] - kyj;
            float sq = dx * dx + dy * dy;
            float dist = (sq == 0.f) ? 0.f : (sqrtf(sq) * MAXD_INV);
            float cc = c[r] + dist * a0r[r] + (1.f - dist) * a1r[r]
                            + dist * c0j   + (1.f - dist) * c1j;

            float rm = cc;
            rm = fmaxf(rm, __shfl_xor(rm, 1, 32));
            rm = fmaxf(rm, __shfl_xor(rm, 2, 32));
            rm = fmaxf(rm, __shfl_xor(rm, 4, 32));
            rm = fmaxf(rm, __shfl_xor(rm, 8, 32));
            float nm = fmaxf(m[r], rm);
            float pe = __expf(cc - nm);

            float ps = pe, pd = pe * dist;
            ps += __shfl_xor(ps, 1, 32);  pd += __shfl_xor(pd, 1, 32);
            ps += __shfl_xor(ps, 2, 32);  pd += __shfl_xor(pd, 2, 32);
            ps += __shfl_xor(ps, 4, 32);  pd += __shfl_xor(pd, 4, 32);
            ps += __shfl_xor(ps, 8, 32);  pd += __shfl_xor(pd, 8, 32);

            float sc = __expf(m[r] - nm);
            l[r]  = l[r]  * sc + ps;
            t0[r] = t0[r] * sc + pd;
            m[r]  = nm;
            #pragma unroll
            for (int ch = 0; ch < 4; ++ch) Oacc[ch][r] *= sc;

            // stage P in A-matrix-friendly layout (wave-local region)
            pT[wave][(r + 8 * hl) * LP + ln] = pe;
        }
        // wave-local LDS RAW: P stores -> P loads (DS ops are in-order per
        // wave, but make the dependency explicit for the scheduler)
        asm volatile("s_wait_dscnt 0" ::: "memory");

        // ---- O += P @ V  (K=16 -> 4 fp32 WMMA steps per 16-wide d chunk) ----
        #pragma unroll
        for (int s = 0; s < 4; ++s) {
            int kk = 4 * s + hl * 2;
            v2f af = *(const v2f*)&pT[wave][ln * LP + kk];  // A[m=ln][k=kk..kk+1]
            #pragma unroll
            for (int ch = 0; ch < 4; ++ch) {
                v2f bf;
                bf.x = vT[cur][kk * LK       + ch * 16 + ln];   // B[k=kk  ][n]
                bf.y = vT[cur][(kk + 1) * LK + ch * 16 + ln];   // B[k=kk+1][n]
                Oacc[ch] = wmma4(af, bf, Oacc[ch]);
            }
        }
        cur ^= 1;
    }

    // ---- finalize: vals = O/l + t0f*Ev0 + (1-t0f)*Ev1 ; write (B,S,H*D) ----
    float ev0[4], ev1[4];
    #pragma unroll
    for (int ch = 0; ch < 4; ++ch) {
        ev0[ch] = Ev[ch * 16 + ln];
        ev1[ch] = Ev[D_ + ch * 16 + ln];
    }
    #pragma unroll
    for (int r = 0; r < 8; ++r) {
        float inv = 1.0f / l[r];
        float t0f = t0[r] * inv;
        int row = qbase + r + 8 * hl;
        float* orow = out + ((size_t)(b * S_ + row) * E_ + h * D_);
        #pragma unroll
        for (int ch = 0; ch < 4; ++ch) {
            orow[ch * 16 + ln] =
                Oacc[ch][r] * inv + t0f * ev0[ch] + (1.f - t0f) * ev1[ch];
        }
    }
}

// ---------------------------------------------------------------------------
// Launcher
// ---------------------------------------------------------------------------
extern "C" void kernel_launch(void* const* d_in, const int* in_sizes, int n_in,
                              void* d_out, int out_size, void* d_ws, size_t ws_size,
                              hipStream_t stream) {
    (void)in_sizes; (void)n_in; (void)out_size; (void)ws_size;
    const float* query = (const float*)d_in[0];
    const float* key_in= (const float*)d_in[1];
    const float* value = (const float*)d_in[2];
    const float* pos   = (const float*)d_in[3];
    const float* Wq    = (const float*)d_in[4];
    const float* bq    = (const float*)d_in[5];
    const float* Wk    = (const float*)d_in[6];
    const float* bk    = (const float*)d_in[7];
    const float* Wv    = (const float*)d_in[8];
    const float* bv    = (const float*)d_in[9];
    const float* Ek    = (const float*)d_in[10];
    const float* Eq    = (const float*)d_in[11];
    const float* Ev    = (const float*)d_in[12];
    float* out = (float*)d_out;

    float* ws = (float*)d_ws;
    const size_t NQKV = (size_t)B_ * S_ * E_;       // 2,097,152 floats each
    const size_t NBHS = (size_t)B_ * H_ * S_;       // 32,768 floats each
    float* qo = ws;
    float* ko = ws + NQKV;
    float* vo = ws + 2 * NQKV;
    float* a0 = ws + 3 * NQKV;
    float* a1 = a0 + NBHS;
    float* c0 = a1 + NBHS;
    float* c1 = c0 + NBHS;

    // 1) Q/K/V projections: 1024 tiles of 32x64 per matrix, 4 waves/block
    dim3 gp(256, 1, 3);
    qkv_proj_kernel<<<gp, 128, 0, stream>>>(query, key_in, value,
                                            Wq, Wk, Wv, bq, bk, bv,
                                            qo, ko, vo);
    // 2) edge-bias dot products
    edge_dots_kernel<<<(B_ * H_ * S_) / 256, 256, 0, stream>>>(
        qo, ko, Ek, Eq, a0, a1, c0, c1);
    // 3) flash attention with distance bias (async double-buffered K/V)
    dim3 ga(S_ / 64, B_ * H_);
    attn_kernel<<<ga, 128, 0, stream>>>(qo, ko, vo, pos,
                                        a0, a1, c0, c1, Ev, out);
}